// NeighborAggregator_26920855011666
// MI455X (gfx1250) — compile-verified
//
#include <hip/hip_runtime.h>
#include <hip/hip_bf16.h>
#include <cstdint>

// ---------------------------------------------------------------------------
// NeighborAggregator on MI455X (gfx1250):
//   vals = adj_values * x[rows, cols]            (E random gathers)
//   A_raw = segment_sum(vals, rows, N)           (global f32 atomics, 32KB dst)
//   alpha = softmax(A_raw)                       (single-WG wave32 reduction)
// Memory-bound (~75 MB total): edge streams staged through LDS with CDNA5
// async global->LDS B128 copies, double buffered on ASYNCcnt.
// ---------------------------------------------------------------------------

#define TILE 1024   // edges per staged tile
#define BLK  256    // threads per block (8 wave32)

__global__ __launch_bounds__(256) void zero_araw(float* __restrict__ a, int n) {
    int i = blockIdx.x * blockDim.x + threadIdx.x;
    if (i < n) a[i] = 0.0f;
}

__device__ __forceinline__ void async_b128(unsigned lds_dst, unsigned long long gsrc) {
    // CDNA5 async copy: 16 bytes from the global address in vaddr to the LDS address in vdst, tracked on ASYNCcnt.
    asm volatile("global_load_async_to_lds_b128 %0, %1, off"
                 :: "v"(lds_dst), "v"(gsrc) : "memory");
}

__global__ __launch_bounds__(BLK) void edge_tiles(const float* __restrict__ x,
                                                  const float* __restrict__ vals,
                                                  const long long* __restrict__ rows,
                                                  const long long* __restrict__ cols,
                                                  float* __restrict__ a_raw,
                                                  int n, long long ntiles) {
    __shared__ __align__(16) float     s_vals[2][TILE];   //  8 KB
    __shared__ __align__(16) long long s_rows[2][TILE];   // 16 KB
    __shared__ __align__(16) long long s_cols[2][TILE];   // 16 KB

    const int tid = threadIdx.x;

    auto issue = [&](int buf, long long tile) {
        const long long base = tile * (long long)TILE;
        // vals: TILE*4B = 4KB -> 1 x B128 per lane
        async_b128((unsigned)(uintptr_t)&s_vals[buf][tid * 4],
                   (unsigned long long)(uintptr_t)(vals + base + tid * 4));
        // rows: TILE*8B = 8KB -> 2 x B128 per lane
        #pragma unroll
        for (int c = tid; c < TILE / 2; c += BLK)
            async_b128((unsigned)(uintptr_t)&s_rows[buf][c * 2],
                       (unsigned long long)(uintptr_t)(rows + base + c * 2));
        // cols: TILE*8B = 8KB -> 2 x B128 per lane
        #pragma unroll
        for (int c = tid; c < TILE / 2; c += BLK)
            async_b128((unsigned)(uintptr_t)&s_cols[buf][c * 2],
                       (unsigned long long)(uintptr_t)(cols + base + c * 2));
        // 5 async ops per lane per tile (uniform across the wave)
    };

    const long long t0 = blockIdx.x;
    const long long stride = gridDim.x;
    if (t0 >= ntiles) return;

    int buf = 0;
    issue(buf, t0);

    for (long long t = t0; t < ntiles; t += stride) {
        const long long nxt = t + stride;
        if (nxt < ntiles) {
            issue(buf ^ 1, nxt);
            // 5 newer ops outstanding; <=5 means current tile's 5 have landed
            asm volatile("s_wait_asynccnt 5" ::: "memory");
        } else {
            asm volatile("s_wait_asynccnt 0" ::: "memory");
        }
        __syncthreads();   // all waves' async data visible

        #pragma unroll
        for (int k = 0; k < TILE / BLK; ++k) {
            const int i = tid + k * BLK;
            const int r = (int)s_rows[buf][i];
            const int c = (int)s_cols[buf][i];
            const float xv = x[(long long)r * (long long)n + (long long)c]; // random gather
            atomicAdd(&a_raw[r], s_vals[buf][i] * xv);                      // no-return f32 atomic
        }
        __syncthreads();   // done reading buf before it is refilled next round
        buf ^= 1;
    }
}

__global__ __launch_bounds__(256) void edge_tail(const float* __restrict__ x,
                                                 const float* __restrict__ vals,
                                                 const long long* __restrict__ rows,
                                                 const long long* __restrict__ cols,
                                                 float* __restrict__ a_raw,
                                                 int n, long long start, long long count) {
    long long i = start + (long long)blockIdx.x * blockDim.x + threadIdx.x;
    if (i < start + count) {
        const int r = (int)rows[i];
        const int c = (int)cols[i];
        atomicAdd(&a_raw[r], vals[i] * x[(long long)r * (long long)n + (long long)c]);
    }
}

__global__ __launch_bounds__(1024) void softmax8k(const float* __restrict__ a,
                                                  float* __restrict__ alpha, int n) {
    __shared__ float red[32];
    __shared__ float bc;
    const int tid = threadIdx.x;
    const int lane = tid & 31;
    const int wave = tid >> 5;
    const int nw = (blockDim.x + 31) >> 5;

    // pass 1: global max
    float m = -__builtin_inff();
    for (int i = tid; i < n; i += blockDim.x) m = fmaxf(m, a[i]);
    #pragma unroll
    for (int off = 16; off; off >>= 1) m = fmaxf(m, __shfl_xor(m, off, 32));
    if (lane == 0) red[wave] = m;
    __syncthreads();
    if (wave == 0) {
        float v = (lane < nw) ? red[lane] : -__builtin_inff();
        #pragma unroll
        for (int off = 16; off; off >>= 1) v = fmaxf(v, __shfl_xor(v, off, 32));
        if (lane == 0) bc = v;
    }
    __syncthreads();
    m = bc;
    __syncthreads();

    // pass 2: sum of exp
    float s = 0.0f;
    for (int i = tid; i < n; i += blockDim.x) s += __expf(a[i] - m);
    #pragma unroll
    for (int off = 16; off; off >>= 1) s += __shfl_xor(s, off, 32);
    if (lane == 0) red[wave] = s;
    __syncthreads();
    if (wave == 0) {
        float v = (lane < nw) ? red[lane] : 0.0f;
        #pragma unroll
        for (int off = 16; off; off >>= 1) v += __shfl_xor(v, off, 32);
        if (lane == 0) bc = v;
    }
    __syncthreads();
    const float inv = 1.0f / bc;

    // pass 3: write alpha
    for (int i = tid; i < n; i += blockDim.x) alpha[i] = __expf(a[i] - m) * inv;
}

extern "C" void kernel_launch(void* const* d_in, const int* in_sizes, int n_in,
                              void* d_out, int out_size, void* d_ws, size_t ws_size,
                              hipStream_t stream) {
    const float*     x  = (const float*)d_in[0];      // (1, N, N) f32
    const float*     av = (const float*)d_in[1];      // (E,) f32
    const long long* ar = (const long long*)d_in[2];  // (E,) i64
    const long long* ac = (const long long*)d_in[3];  // (E,) i64

    const int       n = out_size / 2;                 // 8192
    const long long e = (long long)in_sizes[1];       // 524288

    float* alpha = (float*)d_out;                     // outputs: alpha | A_raw
    float* a_raw = (float*)d_out + n;

    zero_araw<<<(n + 255) / 256, 256, 0, stream>>>(a_raw, n);

    const long long ntiles = e / TILE;
    if (ntiles > 0) {
        int blocks = (int)(ntiles < 128 ? ntiles : 128);  // 4 tiles/block -> deep pipeline
        edge_tiles<<<blocks, BLK, 0, stream>>>(x, av, ar, ac, a_raw, n, ntiles);
    }
    const long long rem = e - ntiles * (long long)TILE;
    if (rem > 0) {
        edge_tail<<<(int)((rem + 255) / 256), 256, 0, stream>>>(x, av, ar, ac, a_raw,
                                                                n, ntiles * (long long)TILE, rem);
    }

    softmax8k<<<1, 1024, 0, stream>>>(a_raw, alpha, n);
}